// GCN_7559142441491
// MI455X (gfx1250) — compile-verified
//
#include <hip/hip_runtime.h>

typedef float v2f __attribute__((ext_vector_type(2)));
typedef float v8f __attribute__((ext_vector_type(8)));

#define C128 128

// ---------- degree count ----------
__global__ __launch_bounds__(256) void k_zero_u32(unsigned* __restrict__ p, int n) {
    int i = blockIdx.x * 256 + threadIdx.x;
    if (i < n) p[i] = 0u;
}

__global__ __launch_bounds__(256) void k_count_deg(const long long* __restrict__ dst,
                                                   unsigned* __restrict__ deg, int E) {
    int e = blockIdx.x * 256 + threadIdx.x;
    if (e < E) atomicAdd(&deg[(int)dst[e]], 1u);   // native global_atomic_add_u32
}

// ---------- two-level exclusive scan of deg -> rowStart (+cursor, +dinv) ----------
__global__ __launch_bounds__(256) void k_blocksum(const unsigned* __restrict__ deg,
                                                  unsigned* __restrict__ bsum, int N) {
    __shared__ unsigned tmp[256];
    int i = blockIdx.x * 256 + threadIdx.x;
    tmp[threadIdx.x] = (i < N) ? deg[i] : 0u;
    __syncthreads();
    for (int off = 128; off > 0; off >>= 1) {
        if (threadIdx.x < off) tmp[threadIdx.x] += tmp[threadIdx.x + off];
        __syncthreads();
    }
    if (threadIdx.x == 0) bsum[blockIdx.x] = tmp[0];
}

// single block; nb <= 256
__global__ __launch_bounds__(256) void k_scan_bsums(const unsigned* __restrict__ bsum,
                                                    unsigned* __restrict__ bscan, int nb) {
    __shared__ unsigned tmp[256];
    int t = threadIdx.x;
    unsigned v = (t < nb) ? bsum[t] : 0u;
    tmp[t] = v;
    __syncthreads();
    for (int off = 1; off < 256; off <<= 1) {
        unsigned add = (t >= off) ? tmp[t - off] : 0u;
        __syncthreads();
        tmp[t] += add;
        __syncthreads();
    }
    if (t < nb) bscan[t] = tmp[t] - v;   // exclusive
}

__global__ __launch_bounds__(256) void k_rowstart(const unsigned* __restrict__ deg,
                                                  const unsigned* __restrict__ bscan,
                                                  unsigned* __restrict__ rowStart,
                                                  unsigned* __restrict__ cursor,
                                                  float* __restrict__ dinv, int N, int E) {
    __shared__ unsigned tmp[256];
    int t = threadIdx.x;
    int i = blockIdx.x * 256 + t;
    unsigned v = (i < N) ? deg[i] : 0u;
    tmp[t] = v;
    __syncthreads();
    for (int off = 1; off < 256; off <<= 1) {
        unsigned add = (t >= off) ? tmp[t - off] : 0u;
        __syncthreads();
        tmp[t] += add;
        __syncthreads();
    }
    if (i < N) {
        unsigned rs = bscan[blockIdx.x] + tmp[t] - v;  // global exclusive prefix
        rowStart[i] = rs;
        cursor[i]   = rs;
        dinv[i]     = rsqrtf((float)v + 1.0f);
        if (i == 0) rowStart[N] = (unsigned)E;
    }
}

// ---------- scatter edges into CSR order (u32 atomics only) ----------
__global__ __launch_bounds__(256) void k_scatter_csr(const long long* __restrict__ src,
                                                     const long long* __restrict__ dst,
                                                     unsigned* __restrict__ cursor,
                                                     unsigned* __restrict__ csrSrc, int E) {
    int e = blockIdx.x * 256 + threadIdx.x;
    if (e >= E) return;
    int d = (int)dst[e];
    unsigned pos = atomicAdd(&cursor[d], 1u);
    csrSrc[pos] = (unsigned)src[e];
}

// ---------- fp32 WMMA GEMM: C[N x 128] = (relu?)A[N x 128] @ W[128 x 128] ----------
template <bool RELU_A>
__global__ __launch_bounds__(256) void k_gemm128_wmma(const float* __restrict__ A,
                                                      const float* __restrict__ W,
                                                      float* __restrict__ C) {
    const int lane = threadIdx.x & 31;
    const int nt   = threadIdx.x >> 5;   // n-tile 0..7
    const int half = lane >> 4;
    const int j    = lane & 15;
    const size_t row16 = (size_t)blockIdx.x * 16;
    const int    col   = nt * 16 + j;

    const float* ap = A + (row16 + (size_t)j) * C128 + 2 * half;
    const float* wp = W + (size_t)(2 * half) * C128 + col;

    v8f acc = {};
    for (int k0 = 0; k0 < C128; k0 += 4) {
        float a0 = ap[k0];
        float a1 = ap[k0 + 1];
        if (RELU_A) { a0 = fmaxf(a0, 0.0f); a1 = fmaxf(a1, 0.0f); }
        v2f a = {a0, a1};
        v2f b = {wp[(size_t)k0 * C128], wp[(size_t)(k0 + 1) * C128]};
        acc = __builtin_amdgcn_wmma_f32_16x16x4_f32(false, a, false, b,
                                                    (short)0, acc, false, false);
    }

    float* cp = C + (row16 + 8 * (size_t)half) * C128 + col;
#pragma unroll
    for (int g = 0; g < 8; ++g) cp[(size_t)g * C128] = acc[g];
}

// ---------- CSR gather-aggregate: one wave per node, no FP atomics ----------
__global__ __launch_bounds__(256) void k_aggregate(const float* __restrict__ h,
                                                   const float* __restrict__ dinv,
                                                   const unsigned* __restrict__ rowStart,
                                                   const unsigned* __restrict__ csrSrc,
                                                   const float* __restrict__ bias,
                                                   float* __restrict__ agg, int N) {
    int node = blockIdx.x * 8 + (threadIdx.x >> 5);
    if (node >= N) return;
    int lane = threadIdx.x & 31;
    int c4   = lane << 2;

    float di = dinv[node];
    int beg = (int)rowStart[node];
    int end = (int)rowStart[node + 1];
    beg = __builtin_amdgcn_readfirstlane(beg);
    end = __builtin_amdgcn_readfirstlane(end);

    // self-loop + bias init: agg = h*(1/deg) + b
    float4 hv = *(const float4*)(h + (size_t)node * C128 + c4);
    float4 bv = *(const float4*)(bias + c4);
    float s = di * di;
    float ax = hv.x * s + bv.x;
    float ay = hv.y * s + bv.y;
    float az = hv.z * s + bv.z;
    float aw = hv.w * s + bv.w;

    for (int i = beg; i < end; ++i) {
        int srcN = (int)csrSrc[i];
        srcN = __builtin_amdgcn_readfirstlane(srcN);     // uniform -> scalar/saddr path
        float norm = di * dinv[srcN];
        float4 v = *(const float4*)(h + (size_t)srcN * C128 + c4);
        ax += v.x * norm;
        ay += v.y * norm;
        az += v.z * norm;
        aw += v.w * norm;
    }

    float4 r; r.x = ax; r.y = ay; r.z = az; r.w = aw;
    *(float4*)(agg + (size_t)node * C128 + c4) = r;
}

// ---------- final head: out = relu(h) @ Wl + bl ----------
__global__ __launch_bounds__(128) void k_final(const float* __restrict__ h,
                                               const float* __restrict__ Wl,
                                               const float* __restrict__ bl,
                                               float* __restrict__ out, int N) {
    __shared__ float sW[C128 * 10];
    __shared__ float sb[10];
    for (int i = threadIdx.x; i < C128 * 10; i += 128) sW[i] = Wl[i];
    if (threadIdx.x < 10) sb[threadIdx.x] = bl[threadIdx.x];
    __syncthreads();

    int node = blockIdx.x * 128 + threadIdx.x;
    if (node >= N) return;

    float acc[10];
#pragma unroll
    for (int k = 0; k < 10; ++k) acc[k] = sb[k];

    const float* hp = h + (size_t)node * C128;
    for (int c = 0; c < C128; ++c) {
        float v = fmaxf(hp[c], 0.0f);
#pragma unroll
        for (int k = 0; k < 10; ++k) acc[k] += v * sW[c * 10 + k];
    }
    float* op = out + (size_t)node * 10;
#pragma unroll
    for (int k = 0; k < 10; ++k) op[k] = acc[k];
}

extern "C" void kernel_launch(void* const* d_in, const int* in_sizes, int n_in,
                              void* d_out, int out_size, void* d_ws, size_t ws_size,
                              hipStream_t stream) {
    const float*     x  = (const float*)d_in[0];
    const long long* ei = (const long long*)d_in[1];   // int64 edge_index [2,E]
    const float*     W1 = (const float*)d_in[2];
    const float*     b1 = (const float*)d_in[3];
    const float*     W2 = (const float*)d_in[4];
    const float*     b2 = (const float*)d_in[5];
    const float*     Wl = (const float*)d_in[6];
    const float*     bl = (const float*)d_in[7];
    float*           out = (float*)d_out;

    const int N = in_sizes[0] / C128;   // 50000 (divisible by 16)
    const int E = in_sizes[1] / 2;      // 800000
    const long long* src = ei;
    const long long* dst = ei + E;

    auto align256 = [](size_t v) { return (v + 255) & ~(size_t)255; };
    char* ws = (char*)d_ws;
    size_t off = 0;
    unsigned* deg      = (unsigned*)(ws + off); off = align256(off + (size_t)N * 4);
    float*    dinv     = (float*)   (ws + off); off = align256(off + (size_t)N * 4);
    unsigned* rowStart = (unsigned*)(ws + off); off = align256(off + (size_t)(N + 1) * 4);
    unsigned* cursor   = (unsigned*)(ws + off); off = align256(off + (size_t)N * 4);
    unsigned* csrSrc   = (unsigned*)(ws + off); off = align256(off + (size_t)E * 4);
    unsigned* bsum     = (unsigned*)(ws + off); off = align256(off + 256 * 4);
    unsigned* bscan    = (unsigned*)(ws + off); off = align256(off + 256 * 4);
    float*    hbuf     = (float*)   (ws + off); off = align256(off + (size_t)N * C128 * 4);
    float*    aggbuf   = (float*)   (ws + off);

    const int nbN  = (N + 255) / 256;   // 196 (<= 256, required by single-block scan)
    const int nbE  = (E + 255) / 256;
    const int mt   = N / 16;            // 3125 m-tiles
    const int nbAg = (N + 7) / 8;

    // CSR build + normalization
    k_zero_u32   <<<nbN, 256, 0, stream>>>(deg, N);
    k_count_deg  <<<nbE, 256, 0, stream>>>(dst, deg, E);
    k_blocksum   <<<nbN, 256, 0, stream>>>(deg, bsum, N);
    k_scan_bsums <<<1,   256, 0, stream>>>(bsum, bscan, nbN);
    k_rowstart   <<<nbN, 256, 0, stream>>>(deg, bscan, rowStart, cursor, dinv, N, E);
    k_scatter_csr<<<nbE, 256, 0, stream>>>(src, dst, cursor, csrSrc, E);

    // layer 1: h = x @ W1 ; agg1 = h/deg + b1 + sum_{e in CSR} norm * h[src]
    k_gemm128_wmma<false><<<mt,   256, 0, stream>>>(x, W1, hbuf);
    k_aggregate          <<<nbAg, 256, 0, stream>>>(hbuf, dinv, rowStart, csrSrc, b1, aggbuf, N);

    // layer 2: h2 = relu(agg1) @ W2 ; agg2 = h2/deg + b2 + sum norm * h2[src]
    k_gemm128_wmma<true> <<<mt,   256, 0, stream>>>(aggbuf, W2, hbuf);
    k_aggregate          <<<nbAg, 256, 0, stream>>>(hbuf, dinv, rowStart, csrSrc, b2, aggbuf, N);

    // head: out = relu(agg2) @ Wl + bl
    k_final<<<(N + 127) / 128, 128, 0, stream>>>(aggbuf, Wl, bl, out, N);
}